// RGCNJointRepresentation_88467736363756
// MI455X (gfx1250) — compile-verified
//
#include <hip/hip_runtime.h>
#include <math.h>

// CDNA5 (gfx1250) wave32 WMMA, f32 16x16x4.
typedef __attribute__((ext_vector_type(2))) float v2f;
typedef __attribute__((ext_vector_type(8))) float v8f;

__device__ __forceinline__ v8f wmma_f32x4(v2f a, v2f b, v8f c) {
  // 8 args: (neg_a, A, neg_b, B, c_mod, C, reuse_a, reuse_b)
  return __builtin_amdgcn_wmma_f32_16x16x4_f32(false, a, false, b, (short)0, c,
                                               false, false);
}

__device__ __forceinline__ void wait_asynccnt0() {
#if __has_builtin(__builtin_amdgcn_s_wait_asynccnt)
  __builtin_amdgcn_s_wait_asynccnt(0);
#else
  asm volatile("s_wait_asynccnt 0x0" ::: "memory");
#endif
}

// Async global -> LDS copy of 16 bytes (CDNA5 GLOBAL_LOAD_ASYNC_TO_LDS_B128,
// GV mode: 64-bit vaddr, LDS byte offset in vdst VGPR). Tracked by ASYNCcnt.
__device__ __forceinline__ void async_copy_b128(void* lds_ptr,
                                                const void* gptr) {
  unsigned lds_off = (unsigned)(unsigned long long)lds_ptr;  // low 32 = LDS off
  unsigned long long ga = (unsigned long long)gptr;
  asm volatile("global_load_async_to_lds_b128 %0, %1, off"
               :
               : "v"(lds_off), "v"(ga)
               : "memory");
}

// ---------------------------------------------------------------- utilities
__global__ void k_zero(float* __restrict__ p, long long n) {
  long long i = (long long)blockIdx.x * blockDim.x + threadIdx.x;
  if (i < n) p[i] = 0.0f;
}

// ------------------------------------------------- layer 1: scalar edge agg
// s[dst] += x[src]; cnt[dst] += 1   (rank-1 trick: x is [N,1])
__global__ void k_edge_scalar(const int* __restrict__ src,
                              const int* __restrict__ dst,
                              const float* __restrict__ x,
                              float* __restrict__ s, float* __restrict__ cnt,
                              int E) {
  int e = blockIdx.x * blockDim.x + threadIdx.x;
  if (e < E) {
    int j = src[e], i = dst[e];
    atomicAdd(&s[i], x[j]);
    atomicAdd(&cnt[i], 1.0f);
  }
}

// --------------------------- layer 2 GEMMs: y = z1@W_rel2, r = z1@W_root2
// One block = one 16-node tile. z1 built on the fly from (x_i, m_i) scalars.
// 8 waves: waves 0-3 -> y columns [0..64), waves 4-7 -> r columns [0..64).
__global__ __launch_bounds__(256) void k_layer2_gemm(
    const float* __restrict__ x, const float* __restrict__ s,
    const float* __restrict__ cnt, const float* __restrict__ Wroot1,
    const float* __restrict__ Wrel1, const float* __restrict__ b1,
    const float* __restrict__ Wroot2, const float* __restrict__ Wrel2,
    float* __restrict__ y, float* __restrict__ r, int N) {
  __shared__ float z1t[16][132];  // padded stride: 132 % 64 = 4 -> no bank dup
  const int tid = threadIdx.x;
  const int i0 = blockIdx.x * 16;

  for (int idx = tid; idx < 16 * 128; idx += 256) {
    int m = idx >> 7, k = idx & 127;
    int node = i0 + m;
    float v = 0.0f;
    if (node < N) {
      float xv = x[node];
      float mv = s[node] / fmaxf(cnt[node], 1.0f);
      v = fmaxf(xv * Wroot1[k] + mv * Wrel1[k] + b1[k], 0.0f);
    }
    z1t[m][k] = v;
  }
  __syncthreads();

  const int wave = tid >> 5, lane = tid & 31;
  const int mrow = lane & 15, half = lane >> 4;
  const float* Wm = (wave < 4) ? Wrel2 : Wroot2;  // [128][64] row-major
  float* outp = (wave < 4) ? y : r;
  const int n = ((wave & 3) << 4) + mrow;  // output column 0..63

  const float* wp = Wm + (2 * half) * 64 + n;  // bump by 4 rows per k-step
  const float* ja = &z1t[mrow][2 * half];
  v8f acc = {};
  for (int k0 = 0; k0 < 128; k0 += 4) {
    v2f a, b;
    a.x = ja[0];
    a.y = ja[1];
    b.x = wp[0];
    b.y = wp[64];
    acc = wmma_f32x4(a, b, acc);
    ja += 4;
    wp += 256;
  }
  for (int v = 0; v < 8; ++v) {
    int node = i0 + v + 8 * half;
    if (node < N) outp[(long long)node * 64 + n] = acc[v];
  }
}

// ------------------------------- layer 2: 64-wide edge scatter (L2 atomics)
// 16 threads per edge, float4 gather of y[src], 4 atomic adds into agg[dst].
__global__ void k_edge_agg64(const int* __restrict__ src,
                             const int* __restrict__ dst,
                             const float* __restrict__ y,
                             float* __restrict__ agg, long long tasks) {
  long long t = (long long)blockIdx.x * blockDim.x + threadIdx.x;
  if (t >= tasks) return;
  int e = (int)(t >> 4);
  int c4 = (int)(t & 15) << 2;
  int j = src[e], i = dst[e];
  float4 v = *(const float4*)(y + (long long)j * 64 + c4);
  float* ap = agg + (long long)i * 64 + c4;
  atomicAdd(ap + 0, v.x);
  atomicAdd(ap + 1, v.y);
  atomicAdd(ap + 2, v.z);
  atomicAdd(ap + 3, v.w);
}

// ------------------------------------ layer 2 combine: z2 (in-place over y)
__global__ void k_layer2_combine(const float* __restrict__ r,
                                 const float* __restrict__ agg,
                                 const float* __restrict__ cnt,
                                 const float* __restrict__ b2,
                                 float* __restrict__ z2, long long total) {
  long long t = (long long)blockIdx.x * blockDim.x + threadIdx.x;
  if (t >= total) return;
  int i = (int)(t >> 6);
  int c = (int)(t & 63);
  float cv = fmaxf(cnt[i], 1.0f);
  z2[t] = fmaxf(r[t] + agg[t] / cv + b2[c], 0.0f);
}

// ------------------------------ decode: fused joint GEMM + MLP head + softmax
// Block = 32 edges (two 16-row M-tiles). joint[32][832] staged in LDS:
// node_rep via VALU, edge_attr via GLOBAL_LOAD_ASYNC_TO_LDS_B128 (ASYNCcnt).
// 8 waves: wave w owns output columns [16w,16w+16); each wave keeps TWO
// accumulators sharing one B-fragment -> 2 wmma per B load.
__global__ __launch_bounds__(256) void k_decode(
    const int* __restrict__ src, const int* __restrict__ dst,
    const float* __restrict__ z2, const float* __restrict__ eattr,
    const float* __restrict__ Wlin1, const float* __restrict__ blin1,
    const float* __restrict__ Wlin2, const float* __restrict__ blin2,
    float* __restrict__ out, int E, int dE) {
  __shared__ int s_src[32], s_dst[32];
  __shared__ float joint[32][836];   // 832 cols, padded (836 % 64 = 4)
  __shared__ float htile[32][132];   // 128 cols, padded
  const int tid = threadIdx.x;
  const int e0 = blockIdx.x * 32;

  if (tid < 32) {
    int e = e0 + tid;
    s_src[tid] = (e < E) ? src[e] : 0;
    s_dst[tid] = (e < E) ? dst[e] : 0;
  }
  __syncthreads();

  // edge_attr -> joint[:, 64:64+dE) : async DMA, bypasses VGPRs entirely.
  {
    const int c4n = dE >> 2;  // float4 chunks per row (192)
    for (int idx = tid; idx < 32 * c4n; idx += 256) {
      int m = idx / c4n;
      int c4 = idx - m * c4n;
      int e = e0 + m;
      if (e < E)
        async_copy_b128(&joint[m][64 + (c4 << 2)],
                        eattr + (long long)e * dE + (c4 << 2));
    }
  }
  // node_rep = z2[src] * z2[dst]  -> joint[:, 0:64)   (z2 is L2-resident)
  for (int idx = tid; idx < 32 * 64; idx += 256) {
    int m = idx >> 6, c = idx & 63;
    float v = 0.0f;
    if (e0 + m < E)
      v = z2[(long long)s_src[m] * 64 + c] * z2[(long long)s_dst[m] * 64 + c];
    joint[m][c] = v;
  }
  wait_asynccnt0();
  __syncthreads();

  const int wave = tid >> 5, lane = tid & 31;
  const int mrow = lane & 15, half = lane >> 4;
  const int n = (wave << 4) + mrow;  // output column 0..127
  const int K = 64 + dE;             // 832

  const float* wp = Wlin1 + (2 * half) * 128 + n;  // W_lin1 hot in L2
  const float* ja = &joint[mrow][2 * half];        // M-tile 0 (rows 0..15)
  const float* jb = &joint[16 + mrow][2 * half];   // M-tile 1 (rows 16..31)
  v8f acc0 = {}, acc1 = {};
  for (int k0 = 0; k0 < K; k0 += 4) {
    v2f b, a0, a1;
    b.x = wp[0];
    b.y = wp[128];
    a0.x = ja[0];
    a0.y = ja[1];
    a1.x = jb[0];
    a1.y = jb[1];
    acc0 = wmma_f32x4(a0, b, acc0);   // one B-frag feeds two M-tiles
    acc1 = wmma_f32x4(a1, b, acc1);
    wp += 512;
    ja += 4;
    jb += 4;
  }
  float bn = blin1[n];
  for (int v = 0; v < 8; ++v) {
    htile[v + 8 * half][n] = fmaxf(acc0[v] + bn, 0.0f);
    htile[16 + v + 8 * half][n] = fmaxf(acc1[v] + bn, 0.0f);
  }
  __syncthreads();

  // tiny [128x5] head + softmax, one thread per edge
  if (tid < 32) {
    int e = e0 + tid;
    if (e < E) {
      float lg[5];
      for (int j = 0; j < 5; ++j) lg[j] = blin2[j];
      for (int k = 0; k < 128; ++k) {
        float hv = htile[tid][k];
        for (int j = 0; j < 5; ++j) lg[j] += hv * Wlin2[k * 5 + j];
      }
      float mx = lg[0];
      for (int j = 1; j < 5; ++j) mx = fmaxf(mx, lg[j]);
      float sum = 0.0f;
      for (int j = 0; j < 5; ++j) {
        lg[j] = __expf(lg[j] - mx);
        sum += lg[j];
      }
      float inv = 1.0f / sum;
      for (int j = 0; j < 5; ++j) out[(long long)e * 5 + j] = lg[j] * inv;
    }
  }
}

// ---------------------------------------------------------------- launcher
extern "C" void kernel_launch(void* const* d_in, const int* in_sizes, int n_in,
                              void* d_out, int out_size, void* d_ws,
                              size_t ws_size, hipStream_t stream) {
  const float* x      = (const float*)d_in[0];
  const int*   tei    = (const int*)d_in[1];   // [2, E_tr]
  const int*   dei    = (const int*)d_in[2];   // [2, E_dec]
  const float* eattr  = (const float*)d_in[3];
  const float* Wroot1 = (const float*)d_in[4];
  const float* Wrel1  = (const float*)d_in[5];
  const float* b1     = (const float*)d_in[6];
  const float* Wroot2 = (const float*)d_in[7];
  const float* Wrel2  = (const float*)d_in[8];
  const float* b2     = (const float*)d_in[9];
  const float* Wlin1  = (const float*)d_in[10];
  const float* blin1  = (const float*)d_in[11];
  const float* Wlin2  = (const float*)d_in[12];
  const float* blin2  = (const float*)d_in[13];

  const int N    = in_sizes[0];       // x is [N,1]
  const int Etr  = in_sizes[1] / 2;
  const int Edec = in_sizes[2] / 2;
  const int dE   = in_sizes[3] / Edec;  // 768

  // workspace layout (floats): s[N] | cnt[N] | y[N*64] | r[N*64] | agg[N*64]
  float* s   = (float*)d_ws;
  float* cnt = s + N;
  float* y   = cnt + N;                 // later reused as z2 (in place)
  float* r   = y + (size_t)N * 64;
  float* agg = r + (size_t)N * 64;

  const long long n64 = (long long)N * 64;
  k_zero<<<(unsigned)((2LL * N + 255) / 256), 256, 0, stream>>>(s, 2LL * N);
  k_zero<<<(unsigned)((n64 + 255) / 256), 256, 0, stream>>>(agg, n64);

  k_edge_scalar<<<(Etr + 255) / 256, 256, 0, stream>>>(tei, tei + Etr, x, s,
                                                       cnt, Etr);

  k_layer2_gemm<<<(N + 15) / 16, 256, 0, stream>>>(
      x, s, cnt, Wroot1, Wrel1, b1, Wroot2, Wrel2, y, r, N);

  const long long tasks = (long long)Etr * 16;
  k_edge_agg64<<<(unsigned)((tasks + 255) / 256), 256, 0, stream>>>(
      tei, tei + Etr, y, agg, tasks);

  k_layer2_combine<<<(unsigned)((n64 + 255) / 256), 256, 0, stream>>>(
      r, agg, cnt, b2, /*z2=*/y, n64);

  k_decode<<<(Edec + 31) / 32, 256, 0, stream>>>(
      dei, dei + Edec, /*z2=*/y, eattr, Wlin1, blin1, Wlin2, blin2,
      (float*)d_out, Edec, dE);
}